// AngleFreqEnhance_51711406243864
// MI455X (gfx1250) — compile-verified
//
#include <hip/hip_runtime.h>
#include <hip/hip_bf16.h>
#include <math.h>

// ---------------- CDNA5 WMMA fp32 16x16x4 helpers ----------------
typedef __attribute__((ext_vector_type(2))) float v2f;
typedef __attribute__((ext_vector_type(8))) float v8f;

__device__ __forceinline__ v8f wmma4(v2f a, v2f b, v8f c) {
    // D(16x16) += A(16x4) * B(4x16), all fp32.
    // (neg_a, A, neg_b, B, c_mod, C, reuse_a, reuse_b); neg must be 0 for f32 per ISA.
    return __builtin_amdgcn_wmma_f32_16x16x4_f32(false, a, false, b, (short)0, c,
                                                 false, false);
}

// Problem constants
#define BN 8
#define CC 256
#define MID 16
#define HH 128
#define WW 128
#define HW (HH * WW)          // 16384
#define NANG 8
#define NRAD 9
#define EPSV 1e-8f
#define PI_F 3.14159265358979323846f

// Workspace layout (floats)
#define OFF_FR   ((size_t)0)                 // 16384   DFT real (symmetric)
#define OFF_FI   ((size_t)16384)             // 16384   DFT imag (symmetric)
#define OFF_GAIN ((size_t)32768)             // 16*HW   unshifted gain
#define OFF_P    ((size_t)294912)            // 8*16*HW x_proj / x_enh
#define OFF_AR   ((size_t)2392064)           // 8*16*HW complex buffer A (re)
#define OFF_AI   ((size_t)4489216)           //         complex buffer A (im)
#define OFF_BR   ((size_t)6586368)           // complex buffer B (re)
#define OFF_BI   ((size_t)8683520)           // complex buffer B (im)

// ---------------- setup: ortho DFT matrix (128x128, symmetric) ----------------
__global__ void k_build_dft(float* __restrict__ FR, float* __restrict__ FI) {
    int idx = blockIdx.x * blockDim.x + threadIdx.x;   // 16384
    if (idx >= HW) return;
    int j = idx >> 7, k = idx & 127;
    int p = (j * k) & 127;                             // exact periodic reduction
    float ang = -2.0f * PI_F * (float)p / 128.0f;
    float s, c;
    __sincosf(ang, &s, &c);
    const float nrm = 0.08838834764831845f;            // 1/sqrt(128)
    FR[idx] = c * nrm;
    FI[idx] = s * nrm;
}

// ---------------- setup: gain[m, k1, k2] in UNSHIFTED freq coords ----------------
__global__ void k_build_gain(const float* __restrict__ fw,  // (16, 8, 9)
                             float* __restrict__ GAIN) {
    int idx = blockIdx.x * blockDim.x + threadIdx.x;   // 16384 pixels
    if (idx >= HW) return;
    int k1 = idx >> 7, k2 = idx & 127;
    int hs = (k1 + 64) & 127, ws = (k2 + 64) & 127;    // shifted coords (fold fftshift)
    float dy = (float)(hs - 64), dx = (float)(ws - 64);
    float r = sqrtf(dy * dy + dx * dx);
    int ridx = (int)floorf(r * 0.125f);
    ridx = ridx > (NRAD - 1) ? (NRAD - 1) : ridx;
    float theta = fmodf(atan2f(dy, dx) + PI_F, PI_F);
    const float delta = PI_F / (float)NANG;
    const float halfw = 0.75f * delta;                 // 1.5*delta/2
    float w[NANG], sum = 0.0f;
    #pragma unroll
    for (int a = 0; a < NANG; ++a) {
        float cen = ((float)a + 0.5f) * delta;
        float dist = fabsf(theta - cen);
        float v = 1.0f - dist / halfw;
        v = (dist < halfw) ? (v > 0.0f ? v : 0.0f) : 0.0f;
        w[a] = v;
        sum += v;
    }
    float inv = 1.0f / (sum + EPSV);
    #pragma unroll
    for (int m = 0; m < MID; ++m) {
        float g = 0.0f;
        #pragma unroll
        for (int a = 0; a < NANG; ++a)
            g += w[a] * inv * fw[m * (NANG * NRAD) + a * NRAD + ridx];
        GAIN[(size_t)m * HW + idx] = g;
    }
}

// ---------------- stage 1: x_proj[b,m,pix] = sum_c w_in[m,c] x[b,c,pix] ----------------
__global__ __launch_bounds__(256) void k_downproj(const float* __restrict__ x,
                                                  const float* __restrict__ w_in,
                                                  float* __restrict__ P) {
    const int lane = threadIdx.x & 31, half = lane >> 4, lm = lane & 15;
    const int wave = (blockIdx.x * blockDim.x + threadIdx.x) >> 5;  // 0..2047
    const int b = wave >> 8;
    const int pix0 = (wave & 255) << 6;                             // 64 pixels / wave
    const float* xb = x + (size_t)b * CC * HW;
    v8f a0 = {}, a1 = {}, a2 = {}, a3 = {};
    for (int k0 = 0; k0 < CC; k0 += 4) {
        const int ka = k0 + 2 * half;
        v2f af; af[0] = w_in[lm * CC + ka]; af[1] = w_in[lm * CC + ka + 1];
        const float* xk = xb + (size_t)ka * HW;
        v2f b0, b1, b2, b3;
        b0[0] = xk[pix0 + lm];          b0[1] = xk[HW + pix0 + lm];
        b1[0] = xk[pix0 + 16 + lm];     b1[1] = xk[HW + pix0 + 16 + lm];
        b2[0] = xk[pix0 + 32 + lm];     b2[1] = xk[HW + pix0 + 32 + lm];
        b3[0] = xk[pix0 + 48 + lm];     b3[1] = xk[HW + pix0 + 48 + lm];
        a0 = wmma4(af, b0, a0);
        a1 = wmma4(af, b1, a1);
        a2 = wmma4(af, b2, a2);
        a3 = wmma4(af, b3, a3);
    }
    float* Pb = P + (size_t)b * MID * HW;
    #pragma unroll
    for (int r = 0; r < 8; ++r) {
        int m = r + 8 * half;
        Pb[(size_t)m * HW + pix0 + lm]      = a0[r];
        Pb[(size_t)m * HW + pix0 + 16 + lm] = a1[r];
        Pb[(size_t)m * HW + pix0 + 32 + lm] = a2[r];
        Pb[(size_t)m * HW + pix0 + 48 + lm] = a3[r];
    }
}

// ---------------- stage 2: row DFT: X1 = P_rows @ F (real in, complex out) ------------
// Rows batched over all images: 16384 rows x 128 cols.
__global__ __launch_bounds__(256) void k_rowdft(const float* __restrict__ P,
                                                const float* __restrict__ FR,
                                                const float* __restrict__ FI,
                                                float* __restrict__ AR,
                                                float* __restrict__ AI) {
    const int lane = threadIdx.x & 31, half = lane >> 4, lm = lane & 15;
    const int wave = (blockIdx.x * blockDim.x + threadIdx.x) >> 5;  // 0..8191
    const int row0 = (wave >> 3) << 4;
    const int col0 = (wave & 7) << 4;
    const float* Arow = P + (size_t)(row0 + lm) * WW;
    v8f accR = {}, accI = {};
    for (int kk = 0; kk < 128; kk += 4) {
        const int ka = kk + 2 * half;
        v2f a;  a[0] = Arow[ka];                  a[1] = Arow[ka + 1];
        v2f br; br[0] = FR[ka * WW + col0 + lm];  br[1] = FR[(ka + 1) * WW + col0 + lm];
        v2f bi; bi[0] = FI[ka * WW + col0 + lm];  bi[1] = FI[(ka + 1) * WW + col0 + lm];
        accR = wmma4(a, br, accR);                // shared A-frag, two accumulators
        accI = wmma4(a, bi, accI);
    }
    #pragma unroll
    for (int r = 0; r < 8; ++r) {
        size_t o = (size_t)(row0 + r + 8 * half) * WW + col0 + lm;
        AR[o] = accR[r];
        AI[o] = accI[r];
    }
}

// ---------------- stage 3/5: column transform: D = (Fr + i*sgn*Fi) @ X (complex) ------
// fwd: S1=-1,S2=+1 ; inv (conj): S1=+1,S2=-1.
// Dr = Fr*Xr + S1*Fi*Xi ; Di = Fr*Xi + S2*Fi*Xr.  Per image (128x128 contiguous).
template <int S1, int S2>
__global__ __launch_bounds__(256) void k_coltrans(const float* __restrict__ Xr,
                                                  const float* __restrict__ Xi,
                                                  const float* __restrict__ FR,
                                                  const float* __restrict__ FI,
                                                  float* __restrict__ Dr,
                                                  float* __restrict__ Di) {
    const int lane = threadIdx.x & 31, half = lane >> 4, lm = lane & 15;
    const int wave = (blockIdx.x * blockDim.x + threadIdx.x) >> 5;  // 0..8191
    const int img = wave >> 6;                                      // 128 images
    const int t = wave & 63;
    const int tr = (t >> 3) << 4, tc = (t & 7) << 4;
    const float* xr = Xr + (size_t)img * HW;
    const float* xi = Xi + (size_t)img * HW;
    v8f accR = {}, accI = {};
    for (int kk = 0; kk < 128; kk += 4) {
        const int ka = kk + 2 * half;
        v2f ar; ar[0] = FR[(tr + lm) * WW + ka]; ar[1] = FR[(tr + lm) * WW + ka + 1];
        v2f ai; ai[0] = FI[(tr + lm) * WW + ka]; ai[1] = FI[(tr + lm) * WW + ka + 1];
        v2f br; br[0] = xr[ka * WW + tc + lm];   br[1] = xr[(ka + 1) * WW + tc + lm];
        v2f bi; bi[0] = xi[ka * WW + tc + lm];   bi[1] = xi[(ka + 1) * WW + tc + lm];
        v2f nai = -ai;                 // f32 WMMA forbids NEG[0:1]; negate via VALU
        accR = wmma4(ar, br, accR);
        accR = wmma4((S1 > 0) ? ai : nai, bi, accR);
        accI = wmma4(ar, bi, accI);
        accI = wmma4((S2 > 0) ? ai : nai, br, accI);
    }
    #pragma unroll
    for (int r = 0; r < 8; ++r) {
        size_t o = (size_t)img * HW + (size_t)(tr + r + 8 * half) * WW + tc + lm;
        Dr[o] = accR[r];
        Di[o] = accI[r];
    }
}

// ---------------- stage 4: gain * (1 + EPS/|X|) in unshifted coords (in-place) --------
__global__ void k_gain_apply(float* __restrict__ Zr, float* __restrict__ Zi,
                             const float* __restrict__ GAIN) {
    size_t idx = (size_t)blockIdx.x * blockDim.x + threadIdx.x;  // 128*16384
    if (idx >= (size_t)BN * MID * HW) return;
    int m = (int)((idx >> 14) & 15);
    int p = (int)(idx & (HW - 1));
    float re = Zr[idx], im = Zi[idx];
    float g = GAIN[(size_t)m * HW + p];
    float r2 = re * re + im * im;
    if (r2 > 0.0f) {
        float sc = g * (1.0f + EPSV * rsqrtf(r2));   // (|X|+EPS)/|X| * gain
        Zr[idx] = sc * re;
        Zi[idx] = sc * im;
    } else {
        Zr[idx] = g * EPSV;                          // theta = atan2(0,0) = 0
        Zi[idx] = 0.0f;
    }
}

// ---------------- stage 6: row IDFT, real part: x_enh = X4r@Fr + X4i@Fi ----------------
__global__ __launch_bounds__(256) void k_rowinv(const float* __restrict__ AR,
                                                const float* __restrict__ AI,
                                                const float* __restrict__ FR,
                                                const float* __restrict__ FI,
                                                float* __restrict__ P) {
    const int lane = threadIdx.x & 31, half = lane >> 4, lm = lane & 15;
    const int wave = (blockIdx.x * blockDim.x + threadIdx.x) >> 5;  // 0..8191
    const int row0 = (wave >> 3) << 4;
    const int col0 = (wave & 7) << 4;
    const float* a1row = AR + (size_t)(row0 + lm) * WW;
    const float* a2row = AI + (size_t)(row0 + lm) * WW;
    v8f acc = {};
    for (int kk = 0; kk < 128; kk += 4) {
        const int ka = kk + 2 * half;
        v2f a1; a1[0] = a1row[ka];                a1[1] = a1row[ka + 1];
        v2f a2; a2[0] = a2row[ka];                a2[1] = a2row[ka + 1];
        v2f b1; b1[0] = FR[ka * WW + col0 + lm];  b1[1] = FR[(ka + 1) * WW + col0 + lm];
        v2f b2; b2[0] = FI[ka * WW + col0 + lm];  b2[1] = FI[(ka + 1) * WW + col0 + lm];
        acc = wmma4(a1, b1, acc);
        acc = wmma4(a2, b2, acc);                 // + X4i * (-(-Fi)) : conj folds to +Fi
    }
    #pragma unroll
    for (int r = 0; r < 8; ++r)
        P[(size_t)(row0 + r + 8 * half) * WW + col0 + lm] = acc[r];
}

// ---------------- stage 7: out = x + w_out(256x16) @ x_enh(16 x pix) -------------------
__global__ __launch_bounds__(256) void k_upproj(const float* __restrict__ x,
                                                const float* __restrict__ w_out,
                                                const float* __restrict__ P,
                                                float* __restrict__ out) {
    const int lane = threadIdx.x & 31, half = lane >> 4, lm = lane & 15;
    const int wave = (blockIdx.x * blockDim.x + threadIdx.x) >> 5;  // 0..8191
    const int b = wave >> 10;
    const int pix0 = (wave & 1023) << 4;
    const float* Pb = P + (size_t)b * MID * HW;
    v2f bf[4];                                     // B-frags for K=16, reused 16x
    #pragma unroll
    for (int s = 0; s < 4; ++s) {
        int ka = 4 * s + 2 * half;
        bf[s][0] = Pb[(size_t)ka * HW + pix0 + lm];
        bf[s][1] = Pb[(size_t)(ka + 1) * HW + pix0 + lm];
    }
    const float* xb = x + (size_t)b * CC * HW;
    float* ob = out + (size_t)b * CC * HW;
    for (int ct = 0; ct < 16; ++ct) {
        v8f acc = {};
        #pragma unroll
        for (int s = 0; s < 4; ++s) {
            int ka = 4 * s + 2 * half;
            v2f a; a[0] = w_out[(ct * 16 + lm) * MID + ka];
            a[1] = w_out[(ct * 16 + lm) * MID + ka + 1];
            acc = wmma4(a, bf[s], acc);
        }
        #pragma unroll
        for (int r = 0; r < 8; ++r) {
            int c = ct * 16 + r + 8 * half;
            size_t o = (size_t)c * HW + pix0 + lm;
            ob[o] = xb[o] + acc[r];
        }
    }
}

extern "C" void kernel_launch(void* const* d_in, const int* in_sizes, int n_in,
                              void* d_out, int out_size, void* d_ws, size_t ws_size,
                              hipStream_t stream) {
    const float* x    = (const float*)d_in[0];   // (8,256,128,128)
    const float* w_in = (const float*)d_in[1];   // (16,256)
    const float* w_out= (const float*)d_in[2];   // (256,16)
    const float* fw   = (const float*)d_in[3];   // (16,8,9)
    float* out = (float*)d_out;
    float* ws  = (float*)d_ws;

    float* FR = ws + OFF_FR;
    float* FI = ws + OFF_FI;
    float* GN = ws + OFF_GAIN;
    float* P  = ws + OFF_P;
    float* AR = ws + OFF_AR;
    float* AI = ws + OFF_AI;
    float* BR = ws + OFF_BR;
    float* BI = ws + OFF_BI;

    k_build_dft<<<64, 256, 0, stream>>>(FR, FI);
    k_build_gain<<<64, 256, 0, stream>>>(fw, GN);
    k_downproj<<<256, 256, 0, stream>>>(x, w_in, P);                  // 2048 waves
    k_rowdft<<<1024, 256, 0, stream>>>(P, FR, FI, AR, AI);            // 8192 waves
    k_coltrans<-1, 1><<<1024, 256, 0, stream>>>(AR, AI, FR, FI, BR, BI);
    k_gain_apply<<<8192, 256, 0, stream>>>(BR, BI, GN);
    k_coltrans<1, -1><<<1024, 256, 0, stream>>>(BR, BI, FR, FI, AR, AI);
    k_rowinv<<<1024, 256, 0, stream>>>(AR, AI, FR, FI, P);
    k_upproj<<<1024, 256, 0, stream>>>(x, w_out, P, out);
}